// InfectionGN_28647431864464
// MI455X (gfx1250) — compile-verified
//
#include <hip/hip_runtime.h>

typedef float v2f __attribute__((ext_vector_type(2)));
typedef float v8f __attribute__((ext_vector_type(8)));

__device__ __forceinline__ float relu_f(float x) { return fmaxf(x, 0.0f); }

// Agent-scope relaxed fp add -> native global_atomic_add_f32 (no CAS loop).
// 32-bit element index from a uniform base => SADDR + voffset (scale_offset).
__device__ __forceinline__ void atomic_add_idx(float* __restrict__ base, int idx, float v) {
    __hip_atomic_fetch_add(base + idx, v, __ATOMIC_RELAXED, __HIP_MEMORY_SCOPE_AGENT);
}

// ---------------------------------------------------------------------------
// Zero-fill for workspace accumulators (stateless per call).
// ---------------------------------------------------------------------------
__global__ void zero_fill_kernel(float* __restrict__ p, size_t n) {
    size_t i = (size_t)blockIdx.x * blockDim.x + threadIdx.x;
    size_t stride = (size_t)gridDim.x * blockDim.x;
    for (; i < n; i += stride) p[i] = 0.0f;
}

// ---------------------------------------------------------------------------
// Kernel A: node encoder  n1 = relu(nf[N,4] @ W[4,8] + b[8])
// ---------------------------------------------------------------------------
__global__ __launch_bounds__(256) void node_encoder_kernel(
    const float* __restrict__ nf, const float* __restrict__ W,
    const float* __restrict__ b, float* __restrict__ n1, int N)
{
    int i = blockIdx.x * blockDim.x + threadIdx.x;
    if (i >= N) return;
    const float4 x = reinterpret_cast<const float4*>(nf)[i];
    float o[8];
    #pragma unroll
    for (int j = 0; j < 8; ++j) {
        float v = b[j] + x.x * W[0 * 8 + j] + x.y * W[1 * 8 + j]
                       + x.z * W[2 * 8 + j] + x.w * W[3 * 8 + j];
        o[j] = relu_f(v);
    }
    float4* dst = reinterpret_cast<float4*>(n1);
    dst[i * 2 + 0] = make_float4(o[0], o[1], o[2], o[3]);
    dst[i * 2 + 1] = make_float4(o[4], o[5], o[6], o[7]);
}

// ---------------------------------------------------------------------------
// Kernel B: fused edge pass. One wave processes 16 edges per tile:
//   e1 = relu(ef@W_e_enc+b_e_enc)                       (VALU, K=2 tiny)
//   e2 = relu(e1@W_e_hid + n1[snd]@W_s_hid + b_e_hid)   (3x V_WMMA_F32_16X16X4_F32)
//   scatter-add e2 into incp[receiver], count into cntf[receiver]
//
// Operand layouts (wave32, f32 16x16x4):
//   A 16x4 : lane L holds M=L&15; K-pair = (L<16)?{0,1}:{2,3} in the 2 VGPRs.
//   B 4x16 : lane L holds N=L&15; same half-selects-upper-K convention.
//   C/D    : VGPR r, lanes 0-15 -> D[r][lane], lanes 16-31 -> D[r+8][lane-16].
// Columns N=8..15 are zero-padded (weights & bias = 0) and never stored.
// All hot-loop addressing uses 32-bit element indices from uniform bases so
// the backend emits SADDR+voffset (scale_offset) forms, not 64-bit VALU math.
// ---------------------------------------------------------------------------
__global__ __launch_bounds__(256) void edge_fused_wmma_kernel(
    const float* __restrict__ ef, const int* __restrict__ snd,
    const int* __restrict__ rcv, const float* __restrict__ n1,
    const float* __restrict__ W_e_enc, const float* __restrict__ b_e_enc,
    const float* __restrict__ W_e_hid, const float* __restrict__ W_s_hid,
    const float* __restrict__ b_e_hid,
    float* __restrict__ incp, float* __restrict__ cntf, int ntiles)
{
    const int lane = threadIdx.x & 31;
    const int hf = lane >> 4;        // lane half: selects upper K pair
    const int nI = lane & 15;        // N for B/C/D, M for A
    const bool act = (nI < 8);

    // B operands (loop-invariant per lane), zero-padded for N>=8
    v2f bE, bS0, bS1;
    {
        const int kE = hf * 2;             // W_e_hid: K = 0..3
        bE.x = act ? W_e_hid[(kE + 0) * 8 + nI] : 0.0f;
        bE.y = act ? W_e_hid[(kE + 1) * 8 + nI] : 0.0f;
        const int kA = hf * 2;             // W_s_hid chunk0: K = 0..3
        const int kB = 4 + hf * 2;         // W_s_hid chunk1: K = 4..7
        bS0.x = act ? W_s_hid[(kA + 0) * 8 + nI] : 0.0f;
        bS0.y = act ? W_s_hid[(kA + 1) * 8 + nI] : 0.0f;
        bS1.x = act ? W_s_hid[(kB + 0) * 8 + nI] : 0.0f;
        bS1.y = act ? W_s_hid[(kB + 1) * 8 + nI] : 0.0f;
    }
    // e1 A-operand constants: this lane produces e1[M=nI][k], k in {hf*2, hf*2+1}
    const int k0 = hf * 2;
    const float we00 = W_e_enc[0 * 4 + k0],     we10 = W_e_enc[1 * 4 + k0],     be0 = b_e_enc[k0];
    const float we01 = W_e_enc[0 * 4 + k0 + 1], we11 = W_e_enc[1 * 4 + k0 + 1], be1 = b_e_enc[k0 + 1];
    const float cb = act ? b_e_hid[nI] : 0.0f;   // bias preload for C

    const float2* __restrict__ ef2 = reinterpret_cast<const float2*>(ef);
    const float2* __restrict__ n1v2 = reinterpret_cast<const float2*>(n1);

    const int wid  = blockIdx.x * (blockDim.x >> 5) + (threadIdx.x >> 5);
    const int wstr = gridDim.x * (blockDim.x >> 5);

    for (int tile = wid; tile < ntiles; tile += wstr) {
        const int base = tile * 16;
        if (tile + wstr < ntiles) {            // prefetch next tile's stream
            const int pb = base + wstr * 16;
            __builtin_prefetch(ef + (size_t)pb * 2, 0, 1);
            __builtin_prefetch(snd + pb, 0, 1);
            __builtin_prefetch(rcv + pb, 0, 1);
        }

        // A operand #1: edge-encoder output pair (computed in VALU, relu'd)
        v2f aE;
        {
            const float2 e = ef2[base + nI];   // 32-bit idx -> saddr+scale form
            aE.x = relu_f(e.x * we00 + e.y * we10 + be0);
            aE.y = relu_f(e.x * we01 + e.y * we11 + be1);
        }
        // A operands #2/#3: gathered sender features (hits L2: n1 is 16MB).
        // Index in float2 units (32-bit) so loads use saddr + scale_offset(8).
        const int s4 = snd[base + nI] * 4;     // row in float2 units
        const float2 g0 = n1v2[s4 + hf];
        const float2 g1 = n1v2[s4 + 2 + hf];
        v2f aS0; aS0.x = g0.x; aS0.y = g0.y;
        v2f aS1; aS1.x = g1.x; aS1.y = g1.y;

        v8f acc = { cb, cb, cb, cb, cb, cb, cb, cb };
        acc = __builtin_amdgcn_wmma_f32_16x16x4_f32(false, aE,  false, bE,  (short)0, acc, false, false);
        acc = __builtin_amdgcn_wmma_f32_16x16x4_f32(false, aS0, false, bS0, (short)0, acc, false, false);
        acc = __builtin_amdgcn_wmma_f32_16x16x4_f32(false, aS1, false, bS1, (short)0, acc, false, false);

        // scatter: lane holds feature nI of edges M = hf*8 + r
        if (act) {
            #pragma unroll
            for (int r = 0; r < 8; ++r) {
                const int m = hf * 8 + r;
                const int d = rcv[base + m];
                atomic_add_idx(incp, d * 8 + nI, relu_f(acc[r]));
            }
        }
        if (lane < 16) {
            atomic_add_idx(cntf, rcv[base + lane], 1.0f);
        }
    }
}

// Scalar fallback for E % 16 edges.
__global__ void edge_remainder_kernel(
    const float* __restrict__ ef, const int* __restrict__ snd,
    const int* __restrict__ rcv, const float* __restrict__ n1,
    const float* __restrict__ W_e_enc, const float* __restrict__ b_e_enc,
    const float* __restrict__ W_e_hid, const float* __restrict__ W_s_hid,
    const float* __restrict__ b_e_hid,
    float* __restrict__ incp, float* __restrict__ cntf, int base, int rem)
{
    int t = blockIdx.x * blockDim.x + threadIdx.x;
    if (t >= rem) return;
    const int e = base + t;
    const float e0 = ef[e * 2 + 0], e1v = ef[e * 2 + 1];
    float h1[4];
    #pragma unroll
    for (int k = 0; k < 4; ++k)
        h1[k] = relu_f(e0 * W_e_enc[k] + e1v * W_e_enc[4 + k] + b_e_enc[k]);
    const int srow = snd[e] * 8;
    const int d = rcv[e];
    #pragma unroll
    for (int j = 0; j < 8; ++j) {
        float v = b_e_hid[j];
        #pragma unroll
        for (int k = 0; k < 4; ++k) v += h1[k] * W_e_hid[k * 8 + j];
        #pragma unroll
        for (int k = 0; k < 8; ++k) v += n1[srow + k] * W_s_hid[k * 8 + j];
        atomic_add_idx(incp, d * 8 + j, relu_f(v));
    }
    atomic_add_idx(cntf, d, 1.0f);
}

// ---------------------------------------------------------------------------
// Kernel C: node hidden block + readouts. One wave = 16 nodes:
//   n2 = relu(n1@W_n_hid + (inc/max(cnt,1))@W_in_hid + b)   (4x WMMA f32 K=4)
//   nodes_out = n2@W_rn + b_rn     (shfl_xor reduction over N lanes)
//   gacc[graph] += n2@W_rg         (f32 atomic, 128 destinations)
// ---------------------------------------------------------------------------
__global__ __launch_bounds__(256) void node_hidden_wmma_kernel(
    const float* __restrict__ n1, const float* __restrict__ incp,
    const float* __restrict__ cntf, const int* __restrict__ n2g,
    const float* __restrict__ W_n_hid, const float* __restrict__ W_in_hid,
    const float* __restrict__ b_n_hid,
    const float* __restrict__ W_rn, const float* __restrict__ b_rn,
    const float* __restrict__ W_rg,
    float* __restrict__ nodes_out, float* __restrict__ gacc, int ntiles)
{
    const int lane = threadIdx.x & 31;
    const int hf = lane >> 4;
    const int nI = lane & 15;
    const bool act = (nI < 8);

    v2f bN0, bN1, bI0, bI1;
    {
        const int kA = hf * 2, kB = 4 + hf * 2;
        bN0.x = act ? W_n_hid[(kA + 0) * 8 + nI] : 0.0f;
        bN0.y = act ? W_n_hid[(kA + 1) * 8 + nI] : 0.0f;
        bN1.x = act ? W_n_hid[(kB + 0) * 8 + nI] : 0.0f;
        bN1.y = act ? W_n_hid[(kB + 1) * 8 + nI] : 0.0f;
        bI0.x = act ? W_in_hid[(kA + 0) * 8 + nI] : 0.0f;
        bI0.y = act ? W_in_hid[(kA + 1) * 8 + nI] : 0.0f;
        bI1.x = act ? W_in_hid[(kB + 0) * 8 + nI] : 0.0f;
        bI1.y = act ? W_in_hid[(kB + 1) * 8 + nI] : 0.0f;
    }
    const float cb  = act ? b_n_hid[nI] : 0.0f;
    const float wrn = act ? W_rn[nI] : 0.0f;
    const float wrg = act ? W_rg[nI] : 0.0f;
    const float brn = b_rn[0];

    const float2* __restrict__ n1v2  = reinterpret_cast<const float2*>(n1);
    const float2* __restrict__ incv2 = reinterpret_cast<const float2*>(incp);

    const int wid  = blockIdx.x * (blockDim.x >> 5) + (threadIdx.x >> 5);
    const int wstr = gridDim.x * (blockDim.x >> 5);

    for (int tile = wid; tile < ntiles; tile += wstr) {
        const int base = tile * 16;
        const int row4 = (base + nI) * 4;      // row in float2 units
        const float2 n0  = n1v2[row4 + hf];
        const float2 n1r = n1v2[row4 + 2 + hf];
        const float sc = 1.0f / fmaxf(cntf[base + nI], 1.0f);
        const float2 i0 = incv2[row4 + hf];
        const float2 i1 = incv2[row4 + 2 + hf];

        v2f aN0; aN0.x = n0.x;      aN0.y = n0.y;
        v2f aN1; aN1.x = n1r.x;     aN1.y = n1r.y;
        v2f aI0; aI0.x = i0.x * sc; aI0.y = i0.y * sc;
        v2f aI1; aI1.x = i1.x * sc; aI1.y = i1.y * sc;

        v8f acc = { cb, cb, cb, cb, cb, cb, cb, cb };
        acc = __builtin_amdgcn_wmma_f32_16x16x4_f32(false, aN0, false, bN0, (short)0, acc, false, false);
        acc = __builtin_amdgcn_wmma_f32_16x16x4_f32(false, aN1, false, bN1, (short)0, acc, false, false);
        acc = __builtin_amdgcn_wmma_f32_16x16x4_f32(false, aI0, false, bI0, (short)0, acc, false, false);
        acc = __builtin_amdgcn_wmma_f32_16x16x4_f32(false, aI1, false, bI1, (short)0, acc, false, false);

        #pragma unroll
        for (int r = 0; r < 8; ++r) {
            const float h = relu_f(acc[r]);          // zero for N>=8 columns
            float tn = h * wrn;
            float tg = h * wrg;
            tn += __shfl_xor(tn, 1, 32);
            tn += __shfl_xor(tn, 2, 32);
            tn += __shfl_xor(tn, 4, 32);
            tn += __shfl_xor(tn, 8, 32);
            tg += __shfl_xor(tg, 1, 32);
            tg += __shfl_xor(tg, 2, 32);
            tg += __shfl_xor(tg, 4, 32);
            tg += __shfl_xor(tg, 8, 32);
            if (nI == 0) {
                const int node = base + hf * 8 + r;
                nodes_out[node] = tn + brn;
                atomic_add_idx(gacc, n2g[node], tg);
            }
        }
    }
}

// Scalar fallback for N % 16 nodes.
__global__ void node_remainder_kernel(
    const float* __restrict__ n1, const float* __restrict__ incp,
    const float* __restrict__ cntf, const int* __restrict__ n2g,
    const float* __restrict__ W_n_hid, const float* __restrict__ W_in_hid,
    const float* __restrict__ b_n_hid,
    const float* __restrict__ W_rn, const float* __restrict__ b_rn,
    const float* __restrict__ W_rg,
    float* __restrict__ nodes_out, float* __restrict__ gacc, int base, int rem)
{
    int t = blockIdx.x * blockDim.x + threadIdx.x;
    if (t >= rem) return;
    const int i = base + t;
    const int row = i * 8;
    const float sc = 1.0f / fmaxf(cntf[i], 1.0f);
    float tn = 0.0f, tg = 0.0f;
    #pragma unroll
    for (int j = 0; j < 8; ++j) {
        float v = b_n_hid[j];
        #pragma unroll
        for (int k = 0; k < 8; ++k)
            v += n1[row + k] * W_n_hid[k * 8 + j] + incp[row + k] * sc * W_in_hid[k * 8 + j];
        v = relu_f(v);
        tn += v * W_rn[j];
        tg += v * W_rg[j];
    }
    nodes_out[i] = tn + b_rn[0];
    atomic_add_idx(gacc, n2g[i], tg);
}

__global__ void finalize_globals_kernel(const float* __restrict__ gacc,
                                        const float* __restrict__ b_rg,
                                        float* __restrict__ out, int B)
{
    int g = blockIdx.x * blockDim.x + threadIdx.x;
    if (g < B) out[g] = gacc[g] + b_rg[0];
}

// ---------------------------------------------------------------------------
extern "C" void kernel_launch(void* const* d_in, const int* in_sizes, int n_in,
                              void* d_out, int out_size, void* d_ws, size_t ws_size,
                              hipStream_t stream) {
    const float* nf       = (const float*)d_in[0];
    const float* ef       = (const float*)d_in[1];
    const int*   snd      = (const int*)d_in[2];
    const int*   rcv      = (const int*)d_in[3];
    const int*   n2g      = (const int*)d_in[4];
    const float* W_e_enc  = (const float*)d_in[5];
    const float* b_e_enc  = (const float*)d_in[6];
    const float* W_n_enc  = (const float*)d_in[7];
    const float* b_n_enc  = (const float*)d_in[8];
    const float* W_e_hid  = (const float*)d_in[9];
    const float* W_s_hid  = (const float*)d_in[10];
    const float* b_e_hid  = (const float*)d_in[11];
    const float* W_n_hid  = (const float*)d_in[12];
    const float* W_in_hid = (const float*)d_in[13];
    const float* b_n_hid  = (const float*)d_in[14];
    const float* W_rn     = (const float*)d_in[15];
    const float* b_rn     = (const float*)d_in[16];
    const float* W_rg     = (const float*)d_in[17];
    const float* b_rg     = (const float*)d_in[18];

    const int N = in_sizes[0] / 4;   // node_features [N,4]
    const int E = in_sizes[1] / 2;   // edge_features [E,2]
    const int B = out_size - N;      // globals count

    // Workspace layout (floats): n1[N*8] | inc[N*8] | cnt[N] | gacc[B]
    float* n1   = (float*)d_ws;
    float* incp = n1 + (size_t)N * 8;
    float* cntf = incp + (size_t)N * 8;
    float* gacc = cntf + N;

    float* nodes_out   = (float*)d_out;
    float* globals_out = nodes_out + N;

    // Zero accumulators (inc, cnt, gacc are contiguous): N*9 + B floats.
    {
        const size_t zn = (size_t)N * 9 + (size_t)B;
        zero_fill_kernel<<<2048, 256, 0, stream>>>(incp, zn);
    }

    // Node encoder
    node_encoder_kernel<<<(N + 255) / 256, 256, 0, stream>>>(nf, W_n_enc, b_n_enc, n1, N);

    // Fused edge pass
    const int etiles = E / 16;
    if (etiles > 0) {
        int needed = (etiles + 7) / 8;            // 8 waves per 256-thread block
        int blocks = needed < 8192 ? needed : 8192;
        edge_fused_wmma_kernel<<<blocks, 256, 0, stream>>>(
            ef, snd, rcv, n1, W_e_enc, b_e_enc, W_e_hid, W_s_hid, b_e_hid,
            incp, cntf, etiles);
    }
    const int erem = E - etiles * 16;
    if (erem > 0) {
        edge_remainder_kernel<<<1, 64, 0, stream>>>(
            ef, snd, rcv, n1, W_e_enc, b_e_enc, W_e_hid, W_s_hid, b_e_hid,
            incp, cntf, etiles * 16, erem);
    }

    // Node hidden + readouts
    const int ntiles = N / 16;
    if (ntiles > 0) {
        int needed = (ntiles + 7) / 8;
        int blocks = needed < 4096 ? needed : 4096;
        node_hidden_wmma_kernel<<<blocks, 256, 0, stream>>>(
            n1, incp, cntf, n2g, W_n_hid, W_in_hid, b_n_hid,
            W_rn, b_rn, W_rg, nodes_out, gacc, ntiles);
    }
    const int nrem = N - ntiles * 16;
    if (nrem > 0) {
        node_remainder_kernel<<<1, 64, 0, stream>>>(
            n1, incp, cntf, n2g, W_n_hid, W_in_hid, b_n_hid,
            W_rn, b_rn, W_rg, nodes_out, gacc, ntiles * 16, nrem);
    }

    // Globals
    finalize_globals_kernel<<<(B + 127) / 128, 128, 0, stream>>>(gacc, b_rg, globals_out, B);
}